// dot_attention_2199023255762
// MI455X (gfx1250) — compile-verified
//
#include <hip/hip_runtime.h>
#include <hip/hip_bf16.h>
#include <cstdint>

typedef __attribute__((ext_vector_type(2))) float v2f;
typedef __attribute__((ext_vector_type(8))) float v8f;

#define BB 2
#define SS 1024
#define HH 8
#define DD 64
#define QT 16            // q rows per block
#define NW 4             // waves per block (wave32)
#define THREADS (NW * 32)

// Row-dependent XOR swizzle of the column index inside the 16x1024 LDS tile.
// Flips bits [5:2] only -> 4-element groups stay contiguous (b64-safe), and
// column-strided reads in the PV phase hit distinct banks per row.
__device__ __forceinline__ int swz(int col, int row) {
    return col ^ ((row & 15) << 2);
}

__global__ __launch_bounds__(THREADS)
void dot_attention_kernel(const float* __restrict__ gate,
                          const float* __restrict__ q,
                          const float* __restrict__ k,
                          const float* __restrict__ v,
                          const float* __restrict__ pair,
                          const unsigned char* __restrict__ mask, // bool: 1 byte
                          float* __restrict__ ctx_out,            // [B,S,H,D]
                          float* __restrict__ attn_out)           // [B,S,S,H]
{
    __shared__ float sL[QT * SS];   // logits -> probs (swizzled per row), 64 KB

    const int bid = blockIdx.x;
    const int qt  = bid & (SS / QT - 1);      // 64 q-tiles
    const int h   = (bid >> 6) & (HH - 1);
    const int b   = bid >> 9;
    const int q0  = qt * QT;

    const int tid  = threadIdx.x;
    const int wave = tid >> 5;
    const int lane = tid & 31;
    const int half = lane >> 4;   // selects K-pair inside a 4-wide chunk
    const int lid  = lane & 15;   // row (A) / col (B,C,D) id

    // ---------------- Phase 1: S = Q K^T, mask, + pair -> LDS ----------------
    // A-fragment layout (16x4 f32): lanes 0-15 hold K=0,1; lanes 16-31 K=2,3.
    v2f qa[DD / 4];
    {
        const float* qrow = q + ((size_t)((b * SS + q0 + lid) * HH + h)) * DD;
        #pragma unroll
        for (int c = 0; c < DD / 4; ++c)
            qa[c] = *(const v2f*)(qrow + 4 * c + 2 * half);
    }

    for (int kt = wave; kt < SS / 16; kt += NW) {
        const int n0 = kt * 16;
        const float* krow = k + ((size_t)((b * SS + n0 + lid) * HH + h)) * DD;
        v8f acc = {0.f, 0.f, 0.f, 0.f, 0.f, 0.f, 0.f, 0.f};
        #pragma unroll
        for (int c = 0; c < DD / 4; ++c) {
            v2f kb = *(const v2f*)(krow + 4 * c + 2 * half);
            acc = __builtin_amdgcn_wmma_f32_16x16x4_f32(
                false, qa[c], false, kb, (short)0, acc, false, false);
        }
        // Epilogue: C/D layout -> element (m = vr + 8*half, n = lid)
        const int col = n0 + lid;
        #pragma unroll
        for (int vr = 0; vr < 8; ++vr) {
            const int m    = vr + 8 * half;
            const int grow = q0 + m;
            const size_t pidx = (size_t)(b * SS + grow) * SS + col;
            float l = acc[vr];
            if (mask[pidx]) l = -__builtin_inff();
            l += pair[pidx * HH + h];
            sL[m * SS + swz(col, m)] = l;
        }
    }
    __syncthreads();

    // ---------------- Phase 2: softmax over key axis, emit attn --------------
    {
        const int row = tid >> 3;            // 0..15
        const int sub = tid & 7;             // 8 threads per row
        const int c0  = sub * (SS / 8);      // 128 contiguous columns each
        float* srow = sL + row * SS;

        float mx = -__builtin_inff();
        for (int c = c0; c < c0 + SS / 8; ++c)
            mx = fmaxf(mx, srow[swz(c, row)]);
        #pragma unroll
        for (int off = 1; off < 8; off <<= 1)
            mx = fmaxf(mx, __shfl_xor(mx, off, 8));

        float sum = 0.f;
        for (int c = c0; c < c0 + SS / 8; ++c) {
            float e = __expf(srow[swz(c, row)] - mx);
            srow[swz(c, row)] = e;
            sum += e;
        }
        #pragma unroll
        for (int off = 1; off < 8; off <<= 1)
            sum += __shfl_xor(sum, off, 8);
        const float inv = 1.0f / sum;

        float* arow = attn_out + (size_t)(b * SS + q0 + row) * SS * HH + h;
        for (int c = c0; c < c0 + SS / 8; ++c) {
            float p = srow[swz(c, row)] * inv;
            srow[swz(c, row)] = p;          // keep normalized probs for PV
            arow[(size_t)c * HH] = p;       // strided; coalesces in 192MB L2
        }
    }
    __syncthreads();

    // ---------------- Phase 3: context = P V, * gate -------------------------
    {
        const int d0 = wave * 16;           // each wave: 16 output columns
        v8f cacc = {0.f, 0.f, 0.f, 0.f, 0.f, 0.f, 0.f, 0.f};
        for (int c = 0; c < SS / 4; ++c) {
            // A from LDS: row = lid, k-pair = 4c + 2*half .. +1 (swizzled)
            const int ci = ((c ^ lid) << 2) + 2 * half;  // == swz(4c+2*half, lid)
            v2f pa = *(const v2f*)(sL + lid * SS + ci);
            // B from V: element (kk, n) = V[4c+kk][d0+n]
            const int kr = 4 * c + 2 * half;
            const float* vbase = v + ((size_t)((b * SS + kr) * HH + h)) * DD + d0 + lid;
            v2f vb = { vbase[0], vbase[(size_t)HH * DD] };
            cacc = __builtin_amdgcn_wmma_f32_16x16x4_f32(
                false, pa, false, vb, (short)0, cacc, false, false);
        }
        #pragma unroll
        for (int vr = 0; vr < 8; ++vr) {
            const int m = vr + 8 * half;
            const size_t idx = ((size_t)((b * SS + q0 + m) * HH + h)) * DD + d0 + lid;
            ctx_out[idx] = cacc[vr] * gate[idx];
        }
    }
}

extern "C" void kernel_launch(void* const* d_in, const int* in_sizes, int n_in,
                              void* d_out, int out_size, void* d_ws, size_t ws_size,
                              hipStream_t stream) {
    (void)in_sizes; (void)n_in; (void)out_size; (void)d_ws; (void)ws_size;
    const float* gate = (const float*)d_in[0];
    const float* q    = (const float*)d_in[1];
    const float* k    = (const float*)d_in[2];
    const float* v    = (const float*)d_in[3];
    const float* pair = (const float*)d_in[4];
    const unsigned char* mask = (const unsigned char*)d_in[5];

    float* ctx  = (float*)d_out;                                   // B*S*H*D
    float* attn = (float*)d_out + (size_t)BB * SS * HH * DD;        // B*S*S*H

    const int grid = BB * HH * (SS / QT);   // 1024 blocks
    dot_attention_kernel<<<grid, THREADS, 0, stream>>>(gate, q, k, v, pair, mask,
                                                       ctx, attn);
}